// BasketballDetector_31748398252158
// MI455X (gfx1250) — compile-verified
//
#include <hip/hip_runtime.h>
#include <hip/hip_bf16.h>

// ---------------- problem constants (from reference) ----------------
#define NB    32            // batch
#define CH    270           // height
#define CW    480           // width
#define HW    (CH * CW)     // 129600 cells per image
#define KTOP  100           // top-K detections
#define PAD   112           // KTOP padded to 7 x 16 WMMA tiles
#define NT    7             // tiles per dim
#define IOU_THRESH 0.7f
#define Q4    (HW / 4)      // 32400 float4 per image

typedef __attribute__((ext_vector_type(2))) float v2f;
typedef __attribute__((ext_vector_type(8))) float v8f;

// =====================================================================
// Kernel 0: zero the per-batch pass-0 histograms (graph-safe init).
// =====================================================================
__global__ __launch_bounds__(256) void bd_init_kernel(unsigned int* __restrict__ ghist) {
  ghist[blockIdx.x * 256 + threadIdx.x] = 0u;
}

// =====================================================================
// Kernel 1: conf = softmax(cls, axis=1)[:,1] = sigmoid(c1 - c0),
// stored as raw float bits (monotone sortable: all values in (0,1)).
// Fuses the radix-select pass-0 (top 8 bits) histogram via LDS
// privatization + one global atomic per bin per block.
// Streaming float4 loads + gfx1250 global_prefetch_b8.
// =====================================================================
__global__ __launch_bounds__(256) void bd_score_kernel(
    const float* __restrict__ cls, unsigned int* __restrict__ keys,
    unsigned int* __restrict__ ghist) {
  __shared__ unsigned int lhist[256];
  const int b = blockIdx.y;
  const int p = blockIdx.x * 256 + threadIdx.x;   // float4 index within batch

  lhist[threadIdx.x] = 0u;
  __syncthreads();

  if (p < Q4) {
    const float4* c0 = reinterpret_cast<const float4*>(cls + (size_t)b * 2 * HW) + p;
    const float4* c1 = reinterpret_cast<const float4*>(cls + (size_t)b * 2 * HW + HW) + p;
    __builtin_prefetch((const char*)c0 + 4096, 0, 1);   // -> global_prefetch_b8
    __builtin_prefetch((const char*)c1 + 4096, 0, 1);

    float4 a = *c0;
    float4 c = *c1;
    uint4 k;
    k.x = __float_as_uint(1.0f / (1.0f + __expf(a.x - c.x)));
    k.y = __float_as_uint(1.0f / (1.0f + __expf(a.y - c.y)));
    k.z = __float_as_uint(1.0f / (1.0f + __expf(a.z - c.z)));
    k.w = __float_as_uint(1.0f / (1.0f + __expf(a.w - c.w)));
    reinterpret_cast<uint4*>(keys + (size_t)b * HW)[p] = k;

    atomicAdd(&lhist[k.x >> 24], 1u);
    atomicAdd(&lhist[k.y >> 24], 1u);
    atomicAdd(&lhist[k.z >> 24], 1u);
    atomicAdd(&lhist[k.w >> 24], 1u);
  }
  __syncthreads();

  unsigned int cnt = lhist[threadIdx.x];
  if (cnt) atomicAdd(&ghist[b * 256 + threadIdx.x], cnt);
}

// =====================================================================
// Kernel 2: exact top-K per batch via MSB radix select; pass 0 comes
// from the precomputed global histogram, passes 1-3 re-scan the keys.
// Candidates (<=256) bitonic-sorted by (score desc, index asc) --
// matches jax.lax.top_k tie-breaking.
// One 1024-thread workgroup (32 wave32) per batch image.
// =====================================================================
__global__ __launch_bounds__(1024) void bd_topk_kernel(
    const unsigned int* __restrict__ keys,
    const unsigned int* __restrict__ ghist,
    int* __restrict__ outIdx, float* __restrict__ outVal) {
  __shared__ unsigned int hist[256];
  __shared__ unsigned int s_prefix, s_need;
  __shared__ unsigned int s_cnt;
  __shared__ unsigned long long cand[256];

  const int b   = blockIdx.x;
  const int tid = threadIdx.x;
  const unsigned int* kb = keys + (size_t)b * HW;

  // ---- pass 0: from precomputed histogram ----
  if (tid < 256) hist[tid] = ghist[b * 256 + tid];
  __syncthreads();
  if (tid == 0) {
    unsigned int cum = 0u, sel = 0u, need0 = KTOP;
    for (int bin = 255; bin >= 0; --bin) {
      unsigned int c = hist[bin];
      if (cum + c >= need0) { sel = (unsigned int)bin; break; }
      cum += c;
    }
    s_prefix = sel << 24;
    s_need   = need0 - cum;
  }
  __syncthreads();
  unsigned int prefix = s_prefix;
  unsigned int need   = s_need;
  __syncthreads();

  // ---- passes 1..3: re-scan keys under resolved prefix ----
  for (int pass = 1; pass < 4; ++pass) {
    const int shift = 24 - pass * 8;
    const unsigned int maskHi = 0xFFFFFFFFu << (shift + 8);

    if (tid < 256) hist[tid] = 0u;
    __syncthreads();

    for (int i = tid; i < HW; i += 1024) {
      unsigned int k = kb[i];
      if ((k & maskHi) == prefix)
        atomicAdd(&hist[(k >> shift) & 0xFFu], 1u);
    }
    __syncthreads();

    if (tid == 0) {
      unsigned int cum = 0u, sel = 0u;
      for (int bin = 255; bin >= 0; --bin) {
        unsigned int c = hist[bin];
        if (cum + c >= need) { sel = (unsigned int)bin; break; }
        cum += c;
      }
      s_prefix = prefix | (sel << shift);
      s_need   = need - cum;
    }
    __syncthreads();
    prefix = s_prefix;
    need   = s_need;
    __syncthreads();                  // protect hist before next-pass zeroing
  }

  const unsigned int T = prefix;      // exact key value of the K-th largest
  if (tid == 0) s_cnt = 0u;
  __syncthreads();

  // Collect all keys >= T (strictly-greater count < K; equals capped at 256).
  for (int i = tid; i < HW; i += 1024) {
    unsigned int k = kb[i];
    if (k >= T) {
      unsigned int pos = atomicAdd(&s_cnt, 1u);
      if (pos < 256u)
        cand[pos] = ((unsigned long long)(~k) << 32) | (unsigned int)i;
    }
  }
  __syncthreads();
  unsigned int total = s_cnt > 256u ? 256u : s_cnt;
  for (int i = tid; i < 256; i += 1024)
    if ((unsigned int)i >= total) cand[i] = ~0ULL;   // sorts last

  // Bitonic sort ascending on ((~key)<<32 | idx) => key desc, idx asc.
  for (int kk = 2; kk <= 256; kk <<= 1) {
    for (int j = kk >> 1; j > 0; j >>= 1) {
      __syncthreads();
      if (tid < 256) {
        int ixj = tid ^ j;
        if (ixj > tid) {
          bool up = ((tid & kk) == 0);
          unsigned long long x = cand[tid], y = cand[ixj];
          if (up ? (x > y) : (x < y)) { cand[tid] = y; cand[ixj] = x; }
        }
      }
    }
  }
  __syncthreads();

  if (tid < KTOP) {
    unsigned long long e = cand[tid];
    unsigned int k = ~(unsigned int)(e >> 32);
    outIdx[b * KTOP + tid] = (int)(e & 0xFFFFFFFFu);
    outVal[b * KTOP + tid] = __uint_as_float(k);
  }
}

// =====================================================================
// Kernel 3: gather-decode boxes + IoU + sequential NMS.
// Pairwise area-sum S[i][j] = area_i + area_j computed as a rank-2
// matmul with V_WMMA_F32_16X16X4_F32 over 7x7 tiles of the padded
// 112x112 matrix; IoU = inter / (S - inter + 1e-9).
// One 128-thread (4 wave32) workgroup per batch image; tile loop
// unrolled x2 to overlap two WMMA+IoU pipelines per wave.
// =====================================================================
__global__ __launch_bounds__(128) void bd_decode_nms_kernel(
    const float* __restrict__ loc,
    const int* __restrict__ topIdx, const float* __restrict__ topVal,
    float* __restrict__ out) {
  __shared__ float sx1[PAD], sy1[PAD], sx2[PAD], sy2[PAD], sarea[PAD];
  __shared__ float siou[PAD * PAD];
  __shared__ int   skeep[PAD];

  const int b = blockIdx.x, tid = threadIdx.x;

  if (tid < PAD) {
    sx1[tid] = 0.f; sy1[tid] = 0.f; sx2[tid] = 0.f; sy2[tid] = 0.f;
    sarea[tid] = 0.f; skeep[tid] = 0;
  }
  __syncthreads();

  if (tid < KTOP) {
    int   idx = topIdx[b * KTOP + tid];
    float val = topVal[b * KTOP + tid];
    float xg = (float)(idx % CW), yg = (float)(idx / CW);
    float xc = xg * 4.0f + 1.5f;           // DS=4, (DS-1)/2 = 1.5
    float yc = yg * 4.0f + 1.5f;
    const float* lb = loc + (size_t)b * 4 * HW;
    float t0 = lb[idx]          * (CW * 4.0f);   // 1920
    float t1 = lb[HW + idx]     * (CH * 4.0f);   // 1080
    float t2 = lb[2 * HW + idx] * (CW * 4.0f);
    float t3 = lb[3 * HW + idx] * (CH * 4.0f);
    float bx = xc + t0, by = yc + t1;
    float x1 = bx - 0.5f * t2, y1 = by - 0.5f * t3;
    float x2 = bx + 0.5f * t2, y2 = by + 0.5f * t3;
    sx1[tid] = x1; sy1[tid] = y1; sx2[tid] = x2; sy2[tid] = y2;
    sarea[tid] = (x2 - x1) * (y2 - y1);
    skeep[tid] = (val >= 0.0f) ? 1 : 0;    // SCORE_THR = 0.0

    float* o = out + ((size_t)b * KTOP + tid) * 5;
    o[0] = x1; o[1] = y1; o[2] = x2; o[3] = y2; o[4] = val;
  }
  __syncthreads();

  // ---- pairwise IoU, 7x7 tiles of 16x16 distributed over 4 waves ----
  const int wave = tid >> 5, lane = tid & 31;
  const int n    = lane & 15;
  const bool lo  = lane < 16;

  #pragma unroll 2
  for (int t = wave; t < NT * NT; t += 4) {        // wave-uniform loop
    int ti = t / NT, tj = t % NT;
    // D = A x B : A(16x4) row m = [area[ti*16+m], 1, 0, 0]
    //             B(4x16) col n = [1, area[tj*16+n], 0, 0]^T
    v8f d;
#if __has_builtin(__builtin_amdgcn_wmma_f32_16x16x4_f32)
    // A layout (32-bit 16x4): lanes 0-15 hold K=0(V0),K=1(V1); lanes 16-31 K=2,3 (zeros)
    v2f av; av.x = lo ? sarea[ti * 16 + n] : 0.f; av.y = lo ? 1.f : 0.f;
    // B layout (4x16): V0: row0 (lanes0-15)/row2 (lanes16-31); V1: row1/row3
    v2f bv; bv.x = lo ? 1.f : 0.f;             bv.y = lo ? sarea[tj * 16 + n] : 0.f;
    v8f cz = (v8f){0.f, 0.f, 0.f, 0.f, 0.f, 0.f, 0.f, 0.f};
    d = __builtin_amdgcn_wmma_f32_16x16x4_f32(
        false, av, false, bv, (short)0, cz, false, false);
#else
    // VALU fallback (same math) so we still compile if the builtin moves
    {
      int rb = ti * 16 + ((lane >> 4) << 3);
      float ac = sarea[tj * 16 + n];
      #pragma unroll
      for (int r = 0; r < 8; ++r) d[r] = sarea[rb + r] + ac;
    }
#endif
    // D layout: VGPR r -> row = r + 8*(lane>=16), col = lane&15
    int col   = tj * 16 + n;
    int rbase = ti * 16 + ((lane >> 4) << 3);
    float cx1 = sx1[col], cy1 = sy1[col], cx2 = sx2[col], cy2 = sy2[col];
    #pragma unroll
    for (int r = 0; r < 8; ++r) {
      int row = rbase + r;
      float xx1 = fmaxf(sx1[row], cx1);
      float yy1 = fmaxf(sy1[row], cy1);
      float xx2 = fminf(sx2[row], cx2);
      float yy2 = fminf(sy2[row], cy2);
      float inter = fmaxf(xx2 - xx1, 0.f) * fmaxf(yy2 - yy1, 0.f);
      siou[row * PAD + col] = inter / (d[r] - inter + 1e-9f);
    }
  }
  __syncthreads();

  // ---- sequential greedy NMS (reference fori_loop semantics) ----
  for (int i = 0; i < KTOP; ++i) {
    if (skeep[i]) {                                  // uniform LDS read
      if (tid < KTOP && tid > i && siou[i * PAD + tid] > IOU_THRESH)
        skeep[tid] = 0;
    }
    __syncthreads();
  }

  if (tid < KTOP)
    out[(size_t)NB * KTOP * 5 + b * KTOP + tid] = skeep[tid] ? 1.0f : 0.0f;
}

// =====================================================================
extern "C" void kernel_launch(void* const* d_in, const int* in_sizes, int n_in,
                              void* d_out, int out_size, void* d_ws, size_t ws_size,
                              hipStream_t stream) {
  const float* cls = (const float*)d_in[0];   // (32, 2, 270, 480) f32
  const float* loc = (const float*)d_in[1];   // (32, 4, 270, 480) f32
  float* out = (float*)d_out;                 // 32*100*5 detections + 32*100 keep

  // workspace layout: keys | per-batch pass-0 histograms | topIdx | topVal
  unsigned int* keys  = (unsigned int*)d_ws;                          // NB*HW
  char* base = (char*)d_ws + (size_t)NB * HW * sizeof(unsigned int);
  unsigned int* ghist = (unsigned int*)base;                          // NB*256
  base += (size_t)NB * 256 * sizeof(unsigned int);
  int*   topIdx = (int*)base;                                         // NB*KTOP
  float* topVal = (float*)(base + (size_t)NB * KTOP * sizeof(int));   // NB*KTOP

  bd_init_kernel<<<NB, 256, 0, stream>>>(ghist);
  dim3 g1((Q4 + 255) / 256, NB);
  bd_score_kernel<<<g1, 256, 0, stream>>>(cls, keys, ghist);
  bd_topk_kernel<<<NB, 1024, 0, stream>>>(keys, ghist, topIdx, topVal);
  bd_decode_nms_kernel<<<NB, 128, 0, stream>>>(loc, topIdx, topVal, out);
}